// ModelNew_70918499991671
// MI455X (gfx1250) — compile-verified
//
#include <hip/hip_runtime.h>

typedef __attribute__((ext_vector_type(2))) float v2f;
typedef __attribute__((ext_vector_type(8))) float v8f;

#define Bsz 4
#define Tsz 2048
#define Hsz 16
#define Ksz 64
#define Vsz 64
#define CHUNK 16
#define NCHUNK (Tsz / CHUNK)

// D = A(16x4, f32) * B(4x16, f32) + C(16x16, f32) on CDNA5 wave32
__device__ __forceinline__ v8f wmma_f32(v2f a, v2f b, v8f c) {
  return __builtin_amdgcn_wmma_f32_16x16x4_f32(
      /*neg_a=*/false, a, /*neg_b=*/false, b,
      /*c_mod=*/(short)0, c, /*reuse_a=*/false, /*reuse_b=*/false);
}

// Contiguous 8B-aligned LDS fragment load -> ds_load_b64 into an aligned pair.
__device__ __forceinline__ v2f ldv2(const float* p) {
  return *(const v2f*)p;
}

__global__ __launch_bounds__(128) void kda_chunk_wmma_kernel(
    const float* __restrict__ q, const float* __restrict__ k,
    const float* __restrict__ v, const float* __restrict__ g,
    const float* __restrict__ beta, float* __restrict__ out) {
  // ---- LDS ----
  __shared__ float Ssh[Ksz][Vsz + 1];   // recurrent state S (K x V), padded
  __shared__ float qb[CHUNK][Ksz];      // q * exp(G) * scale
  __shared__ float kb[CHUNK][Ksz];      // k * exp(G)            (k-bar)
  __shared__ float wv[CHUNK][Ksz];      // beta * k * exp(-G)    (w)
  __shared__ float wp[CHUNK][Ksz];      // beta * k * exp(GC-G)  (w')
  __shared__ float vv[CHUNK][Vsz];      // values
  __shared__ float DV[CHUNK][Vsz];      // delta-rule corrected values
  __shared__ float Msh[CHUNK][CHUNK];   // strict lower K-bar W^T
  __shared__ float Psh[CHUNK][CHUNK];   // tril Q-bar W^T
  __shared__ float decayC[Ksz];         // exp(G_C) per key channel
  __shared__ float bet[CHUNK];

  const int tid  = threadIdx.x;
  const int lane = tid & 31;
  const int wid  = tid >> 5;   // wave id 0..3 -> V-column stripe
  const int half = lane >> 4;  // lane half (0/1)
  const int mn   = lane & 15;  // row (A) / col (B,C,D) index within 16
  const int coff = wid * 16;   // this wave's V-column offset
  const int h2   = 2 * half;

  const int bh = blockIdx.x;
  const int b  = bh / Hsz;
  const int h  = bh % Hsz;
  const float scale = 0.125f;  // K^-0.5 = 64^-0.5

  // zero the state
  for (int i = tid; i < Ksz * Vsz; i += 128) Ssh[i >> 6][i & 63] = 0.f;
  __syncthreads();

  for (int ch = 0; ch < NCHUNK; ++ch) {
    const int t0 = ch * CHUNK;

    // ---- 1) cooperative load of the chunk (coalesced) ----
    for (int e = tid; e < CHUNK * Ksz; e += 128) {
      const int i = e >> 6, c = e & 63;
      const size_t gidx = ((size_t)((b * Tsz + t0 + i) * Hsz + h)) * Ksz + c;
      qb[i][c] = q[gidx];
      kb[i][c] = k[gidx];
      vv[i][c] = v[gidx];
      if (ch + 1 < NCHUNK && (c & 31) == 0) {  // prefetch next chunk, 1 per 128B line
        const size_t nidx = gidx + (size_t)CHUNK * Hsz * Ksz;
        __builtin_prefetch(&q[nidx], 0, 1);
        __builtin_prefetch(&k[nidx], 0, 1);
        __builtin_prefetch(&v[nidx], 0, 1);
        __builtin_prefetch(&g[nidx], 0, 1);
      }
    }
    if (tid < CHUNK) bet[tid] = beta[(b * Tsz + t0 + tid) * Hsz + h];
    __syncthreads();

    // ---- 2) per-key-channel cumulative decay + decayed operand precompute ----
    // g is read straight from global (coalesced across the 64 channels).
    if (tid < Ksz) {
      const int c = tid;
      float G[CHUNK];
      float acc = 0.f;
      for (int i = 0; i < CHUNK; ++i) {
        acc += g[((size_t)((b * Tsz + t0 + i) * Hsz + h)) * Ksz + c];
        G[i] = acc;
      }
      const float GC = acc;
      decayC[c] = __expf(GC);
      for (int i = 0; i < CHUNK; ++i) {
        const float eG = __expf(G[i]);
        const float kr = kb[i][c];
        const float bi = bet[i];
        kb[i][c] = kr * eG;
        qb[i][c] = qb[i][c] * eG * scale;
        wv[i][c] = bi * kr * __expf(-G[i]);
        wp[i][c] = bi * kr * __expf(GC - G[i]);
      }
    }
    __syncthreads();

    // ---- 3) fused state reads: RHS = V - Kbar*S0 and partial O = Qbar*S0 ----
    // Both chains share the same B fragments of the state stripe: load once,
    // feed two WMMAs.  accO stays live in registers across the barriers.
    v8f accO = {};
    {
      v8f accR = {};
      for (int kk = 0; kk < Ksz; kk += 4) {
        v2f bb = {Ssh[kk + h2][coff + mn], Ssh[kk + h2 + 1][coff + mn]};
        v2f ak = ldv2(&kb[mn][kk + h2]);
        v2f aq = ldv2(&qb[mn][kk + h2]);
        accR = wmma_f32(ak, bb, accR);
        accO = wmma_f32(aq, bb, accO);
      }
      for (int r = 0; r < 8; ++r) {
        const int row = r + 8 * half;
        DV[row][coff + mn] = vv[row][coff + mn] - accR[r];
      }
    }

    // ---- 3b) wave0: M = strict_tril(Kbar W^T); wave1: P = tril(Qbar W^T) ----
    if (wid < 2) {
      const float(*A)[Ksz] = (wid == 0) ? kb : qb;
      v8f acc = {};
      for (int kk = 0; kk < Ksz; kk += 4) {
        v2f a  = ldv2(&A[mn][kk + h2]);
        v2f bt = ldv2(&wv[mn][kk + h2]);  // (W^T) fragment
        acc = wmma_f32(a, bt, acc);
      }
      for (int r = 0; r < 8; ++r) {
        const int row = r + 8 * half;
        const float val = acc[r];
        if (wid == 0) Msh[row][mn] = (mn < row) ? val : 0.f;
        else          Psh[row][mn] = (mn <= row) ? val : 0.f;
      }
    }
    __syncthreads();

    // ---- 4) forward substitution: (I + M) DV = RHS, per value-column ----
    if (tid < Vsz) {
      const int c = tid;
      for (int i = 1; i < CHUNK; ++i) {
        float acc = DV[i][c];
        for (int j = 0; j < i; ++j) acc -= Msh[i][j] * DV[j][c];
        DV[i][c] = acc;
      }
    }
    __syncthreads();

    // ---- 5) finish output: O = (Qbar*S0) + P*DV, store to global ----
    {
      for (int cc = 0; cc < CHUNK; cc += 4) {
        v2f a  = ldv2(&Psh[mn][cc + h2]);
        v2f bb = {DV[cc + h2][coff + mn], DV[cc + h2 + 1][coff + mn]};
        accO = wmma_f32(a, bb, accO);
      }
      for (int r = 0; r < 8; ++r) {
        const int row = r + 8 * half;
        const size_t gidx =
            ((size_t)((b * Tsz + t0 + row) * Hsz + h)) * Vsz + coff + mn;
        out[gidx] = accO[r];
      }
    }

    // ---- 6) state update stripe: S = diag(exp(GC)) S + W'^T DV ----
    for (int kt = 0; kt < 4; ++kt) {
      v8f acc;
      for (int r = 0; r < 8; ++r) {
        const int kg = kt * 16 + r + 8 * half;
        acc[r] = Ssh[kg][coff + mn] * decayC[kg];
      }
      for (int cc = 0; cc < CHUNK; cc += 4) {
        // A[m][c] = W'(time c, key kt*16+m): rows differ -> scalar pair
        v2f a  = {wp[cc + h2][kt * 16 + mn], wp[cc + h2 + 1][kt * 16 + mn]};
        v2f bb = {DV[cc + h2][coff + mn], DV[cc + h2 + 1][coff + mn]};
        acc = wmma_f32(a, bb, acc);
      }
      for (int r = 0; r < 8; ++r) {
        const int kg = kt * 16 + r + 8 * half;
        Ssh[kg][coff + mn] = acc[r];
      }
    }
    __syncthreads();  // protect shared chunk buffers before next load
  }
}

extern "C" void kernel_launch(void* const* d_in, const int* in_sizes, int n_in,
                              void* d_out, int out_size, void* d_ws, size_t ws_size,
                              hipStream_t stream) {
  (void)in_sizes; (void)n_in; (void)out_size; (void)d_ws; (void)ws_size;
  const float* q    = (const float*)d_in[0];
  const float* k    = (const float*)d_in[1];
  const float* v    = (const float*)d_in[2];
  const float* g    = (const float*)d_in[3];
  const float* beta = (const float*)d_in[4];
  float* out = (float*)d_out;

  dim3 grid(Bsz * Hsz);   // one workgroup per (batch, head)
  dim3 block(128);        // 4 wave32s; each owns a 16-wide V stripe
  hipLaunchKernelGGL(kda_chunk_wmma_kernel, grid, block, 0, stream,
                     q, k, v, g, beta, out);
}